// GCNConv_50886772523358
// MI455X (gfx1250) — compile-verified
//
#include <hip/hip_runtime.h>
#include <hip/hip_bf16.h>

typedef __attribute__((ext_vector_type(2))) float v2f;
typedef __attribute__((ext_vector_type(8))) float v8f;

#define CIN  128
#define COUT 128
#define XPAD 132   // LDS row pitch (floats) to avoid bank conflicts on strided A reads

// ---------------------------------------------------------------------------
// Kernel 1: h[i,:] = (x[i,:] @ W) * out_norm[i],  out_norm = 1/sqrt(colptr deg)
// 8 waves/block; wave w owns columns [16w, 16w+16). B fragments for the whole
// K=128 are register-resident per wave (reused across row tiles).
// ---------------------------------------------------------------------------
__global__ __launch_bounds__(256) void gcn_gemm_norm(
    const float* __restrict__ x, const float* __restrict__ w,
    const int* __restrict__ colptr, float* __restrict__ h, int n)
{
  __shared__ float xs[16 * XPAD];
  __shared__ float onorm[16];

  const int tid  = threadIdx.x;
  const int wave = tid >> 5;        // 0..7 -> column tile
  const int lane = tid & 31;
  const int half = lane >> 4;       // 0: lanes 0-15, 1: lanes 16-31
  const int l16  = lane & 15;
  const int n0   = wave * 16;

  // Preload B fragments (W is K-major: w[k*COUT + n]).
  // ISA A/B 16x4 layout: VGPR0 = K {0|2}, VGPR1 = K {1|3} per half-wave.
  v2f bfrag[32];
#pragma unroll
  for (int s = 0; s < 32; ++s) {
    const int kk = 4 * s + 2 * half;
    bfrag[s].x = w[kk       * COUT + n0 + l16];
    bfrag[s].y = w[(kk + 1) * COUT + n0 + l16];
  }

  const int ntiles = (n + 15) >> 4;
  for (int t = blockIdx.x; t < ntiles; t += gridDim.x) {
    const int row0 = t << 4;

    __syncthreads();  // protect xs against previous iteration's readers
    // Cooperative, coalesced load of the 16x128 x tile into padded LDS.
#pragma unroll
    for (int i = 0; i < 8; ++i) {
      const int e = tid + i * 256;          // 0..2047
      const int r = e >> 7, c = e & 127;
      const int gr = row0 + r;
      xs[r * XPAD + c] = (gr < n) ? x[(size_t)gr * CIN + c] : 0.0f;
    }
    if (tid < 16) {
      const int gr = row0 + tid;
      const float deg = (gr < n) ? (float)(colptr[gr + 1] - colptr[gr]) : 1.0f;
      onorm[tid] = rsqrtf(deg);
    }
    __syncthreads();

    v8f acc = {};
#pragma unroll
    for (int s = 0; s < 32; ++s) {
      const int kk = 4 * s + 2 * half;
      v2f a;
      a.x = xs[l16 * XPAD + kk];
      a.y = xs[l16 * XPAD + kk + 1];
      acc = __builtin_amdgcn_wmma_f32_16x16x4_f32(
          /*neg_a=*/false, a, /*neg_b=*/false, bfrag[s],
          /*c_mod=*/(short)0, acc, /*reuse_a=*/false, /*reuse_b=*/false);
    }

    // D layout: VGPR r -> row r (lanes 0-15) / row r+8 (lanes 16-31), col = l16.
#pragma unroll
    for (int r = 0; r < 8; ++r) {
      const int m  = r + 8 * half;
      const int gr = row0 + m;
      if (gr < n)
        h[(size_t)gr * COUT + n0 + l16] = acc[r] * onorm[m];
    }
  }
}

// ---------------------------------------------------------------------------
// Kernel 2: out[i,:] = in_norm[i] * sum_{e in [rowptr[i],rowptr[i+1])} h[colind[e],:] + bias
// One wave per node; lane l handles channels 4l..4l+3 (float4 => 512B/wave per
// neighbor, coalesced, L2-resident since h is 51MB < 192MB L2).
// ---------------------------------------------------------------------------
__global__ __launch_bounds__(256) void gcn_aggr(
    const float* __restrict__ h, const int* __restrict__ rowptr,
    const int* __restrict__ colind, const float* __restrict__ bias,
    float* __restrict__ out, int n)
{
  const int wave = threadIdx.x >> 5;
  const int lane = threadIdx.x & 31;
  const int wavesPerBlk = blockDim.x >> 5;

  const float4* __restrict__ h4 = (const float4*)h;
  const float4  b4 = ((const float4*)bias)[lane];

  for (int i = blockIdx.x * wavesPerBlk + wave; i < n;
       i += gridDim.x * wavesPerBlk) {
    const int e0 = rowptr[i];
    const int e1 = rowptr[i + 1];

    float ax = 0.f, ay = 0.f, az = 0.f, aw = 0.f;
    for (int e = e0; e < e1; ++e) {
      const int nbr = colind[e];
      const float4 v = h4[(size_t)nbr * (COUT / 4) + lane];
      ax += v.x; ay += v.y; az += v.z; aw += v.w;
    }

    const float inorm = rsqrtf((float)(e1 - e0));
    float4 o;
    o.x = ax * inorm + b4.x;
    o.y = ay * inorm + b4.y;
    o.z = az * inorm + b4.z;
    o.w = aw * inorm + b4.w;
    ((float4*)out)[(size_t)i * (COUT / 4) + lane] = o;
  }
}

// ---------------------------------------------------------------------------
extern "C" void kernel_launch(void* const* d_in, const int* in_sizes, int n_in,
                              void* d_out, int out_size, void* d_ws, size_t ws_size,
                              hipStream_t stream) {
  const float* x      = (const float*)d_in[0];   // [N, 128]
  const float* weight = (const float*)d_in[1];   // [128, 128]
  const float* bias   = (const float*)d_in[2];   // [128]
  const int*   rowptr = (const int*)d_in[3];     // [N+1]
  const int*   colind = (const int*)d_in[4];     // [E]
  const int*   colptr = (const int*)d_in[5];     // [N+1]
  // d_in[6] = rowind : unused by the reference computation

  const int n = in_sizes[3] - 1;                 // N from rowptr length
  float* h = (float*)d_ws;                       // [N, 128] fp32 scratch

  dim3 blk(256);
  gcn_gemm_norm<<<1024, blk, 0, stream>>>(x, weight, colptr, h, n);
  gcn_aggr<<<4096, blk, 0, stream>>>(h, rowptr, colind, bias, (float*)d_out, n);
}